// QPIELayer_9603546874291
// MI455X (gfx1250) — compile-verified
//
#include <hip/hip_runtime.h>
#include <math.h>

// ---------------------------------------------------------------------------
// 16-qubit statevector simulator for the QPIE reference circuit, MI455X.
// One workgroup (1024 threads = 32 wave32) per batch element; state lives in
// d_ws (16 x 512KB = 8MB, fully L2-resident on the 192MB L2).
// Dense single-qubit gates run through V_WMMA_F32_16X16X4_F32: the real 4x4
// embedding of the 2x2 complex gate is placed block-diagonally (x4) in a
// 16x16 A operand; 4 chained WMMAs (K=16) transform 64 amplitude pairs per
// wave step. Diagonal gates (CRZ, RZZ) are fused into pointwise phase passes;
// RXX*RYY*RZZ collapses analytically into two 2x2 subspace rotations.
// ---------------------------------------------------------------------------

typedef __attribute__((ext_vector_type(2))) float v2f;
typedef __attribute__((ext_vector_type(8))) float v8f;

#define NQ 8
#define NL 3
#define NA 2
#define NSTATE 65536u   // 2^16 amplitudes
#define BLOCK 1024      // 32 waves -> one full WGP per batch element

#if __has_builtin(__builtin_amdgcn_wmma_f32_16x16x4_f32)
#define HAVE_WMMA4 1
#else
#define HAVE_WMMA4 0
#endif

// insert a 0 bit at position b (shifting higher bits up)
__device__ __forceinline__ unsigned insb(unsigned v, int b) {
  unsigned lo = v & ((1u << b) - 1u);
  return ((v >> b) << (b + 1)) | lo;
}

// workgroup-wide sync with global-memory visibility (state is in global/L2)
__device__ __forceinline__ void wgsync() {
  __threadfence_block();
  __syncthreads();
}

// ---------------------------------------------------------------------------
// Dense single-qubit gate: new = U * old over all 32768 amplitude pairs.
// ---------------------------------------------------------------------------

#if HAVE_WMMA4
// Wave-cooperative WMMA path. Per step a wave transforms 64 pairs:
//  A (16x16, K-chunked by 4) = blockdiag(G4,G4,G4,G4), G4 = real embedding of U
//  B column n, row 4g+c       = component c of pair (base + 4n + g)
//  D row  M=4g+c, col n       = output component c of pair (base + 4n + g)
// ISA f32 layouts: A lane holds M=lane&15, K=(lane>>4)*2 + vgpr; B symmetric
// with N=lane&15; C/D vgpr r: M = r + 8*(lane>>4), N = lane&15.
__device__ __forceinline__ void apply1q(float2* __restrict__ S, int bit,
    float u00r, float u00i, float u01r, float u01i,
    float u10r, float u10i, float u11r, float u11i)
{
  const int lane = threadIdx.x & 31;
  const int wv   = threadIdx.x >> 5;
  const int n    = lane & 15;
  const int hi   = lane >> 4;
  const int c    = n & 3;     // component row within 4-block
  const int g    = n >> 2;    // which 4x4 diagonal block this lane's row is in

  // G4 = [[u00r,-u00i,u01r,-u01i],[u00i,u00r,u01i,u01r],
  //       [u10r,-u10i,u11r,-u11i],[u10i,u10r,u11i,u11r]]
  // branchless (keeps EXEC all-ones around the WMMAs)
  float e0 = (c == 0) ? u00r  : (c == 1) ? u00i : (c == 2) ? u10r  : u10i;
  float e1 = (c == 0) ? -u00i : (c == 1) ? u00r : (c == 2) ? -u10i : u10r;
  float e2 = (c == 0) ? u01r  : (c == 1) ? u01i : (c == 2) ? u11r  : u11i;
  float e3 = (c == 0) ? -u01i : (c == 1) ? u01r : (c == 2) ? -u11i : u11r;
  float col0 = hi ? e2 : e0;  // A column hi*2
  float col1 = hi ? e3 : e1;  // A column hi*2+1

  v2f a0 = { (g == 0) ? col0 : 0.f, (g == 0) ? col1 : 0.f };
  v2f a1 = { (g == 1) ? col0 : 0.f, (g == 1) ? col1 : 0.f };
  v2f a2 = { (g == 2) ? col0 : 0.f, (g == 2) ? col1 : 0.f };
  v2f a3 = { (g == 3) ? col0 : 0.f, (g == 3) ? col1 : 0.f };

  const unsigned m = 1u << bit;
  const unsigned sel = hi ? m : 0u;

#pragma unroll 1
  for (int s = 0; s < 16; ++s) {             // 16 steps * 32 waves * 64 pairs = 32768
    unsigned base = (unsigned)(s * 32 + wv) * 64u;
    v2f b0, b1, b2, b3;
    {
      unsigned i = insb(base + n * 4 + 0, bit) + sel;
      float2 v = S[i]; b0 = (v2f){ v.x, v.y };
    }
    {
      unsigned i = insb(base + n * 4 + 1, bit) + sel;
      float2 v = S[i]; b1 = (v2f){ v.x, v.y };
    }
    {
      unsigned i = insb(base + n * 4 + 2, bit) + sel;
      float2 v = S[i]; b2 = (v2f){ v.x, v.y };
    }
    {
      unsigned i = insb(base + n * 4 + 3, bit) + sel;
      float2 v = S[i]; b3 = (v2f){ v.x, v.y };
    }
    v8f acc = { 0.f, 0.f, 0.f, 0.f, 0.f, 0.f, 0.f, 0.f };
    acc = __builtin_amdgcn_wmma_f32_16x16x4_f32(false, a0, false, b0, (short)0, acc, false, false);
    acc = __builtin_amdgcn_wmma_f32_16x16x4_f32(false, a1, false, b1, (short)0, acc, false, false);
    acc = __builtin_amdgcn_wmma_f32_16x16x4_f32(false, a2, false, b2, (short)0, acc, false, false);
    acc = __builtin_amdgcn_wmma_f32_16x16x4_f32(false, a3, false, b3, (short)0, acc, false, false);
    // lane holds full output pairs  base + 4n + 2*hi + {0,1}
    {
      unsigned i0 = insb(base + n * 4 + hi * 2 + 0, bit);
      S[i0]     = make_float2(acc[0], acc[1]);
      S[i0 + m] = make_float2(acc[2], acc[3]);
    }
    {
      unsigned i0 = insb(base + n * 4 + hi * 2 + 1, bit);
      S[i0]     = make_float2(acc[4], acc[5]);
      S[i0 + m] = make_float2(acc[6], acc[7]);
    }
  }
}
#else
// VALU fallback (only if the f32 WMMA builtin is unavailable)
__device__ __forceinline__ void apply1q(float2* __restrict__ S, int bit,
    float u00r, float u00i, float u01r, float u01i,
    float u10r, float u10i, float u11r, float u11i)
{
  const unsigned m = 1u << bit;
  for (unsigned P = threadIdx.x; P < 32768u; P += BLOCK) {
    unsigned i0 = insb(P, bit), i1 = i0 | m;
    float2 s0 = S[i0], s1 = S[i1];
    S[i0] = make_float2(u00r * s0.x - u00i * s0.y + u01r * s1.x - u01i * s1.y,
                        u00i * s0.x + u00r * s0.y + u01i * s1.x + u01r * s1.y);
    S[i1] = make_float2(u10r * s0.x - u10i * s0.y + u11r * s1.x - u11i * s1.y,
                        u10i * s0.x + u10r * s0.y + u11i * s1.x + u11r * s1.y);
  }
}
#endif

// ---------------------------------------------------------------------------
// Fused RXX(a)*RYY(b)*RZZ(c): block 2x2 rotations on {00,11} (angle (a-b)/2,
// phase e^{-ic/2}) and {01,10} (angle (a+b)/2, phase e^{+ic/2}).
// ---------------------------------------------------------------------------
__device__ __forceinline__ void applyXXYYZZ(float2* __restrict__ S, int q1, int q2,
                                            float a, float b, float c)
{
  int bl = q1 < q2 ? q1 : q2;
  int bh = q1 < q2 ? q2 : q1;
  unsigned ml = 1u << bl, mh = 1u << bh;
  float sd, cd; __sincosf(0.5f * (a - b), &sd, &cd);
  float ss, cs; __sincosf(0.5f * (a + b), &ss, &cs);
  float sc, cc; __sincosf(0.5f * c, &sc, &cc);
  for (unsigned k = threadIdx.x; k < 16384u; k += BLOCK) {
    unsigned b00 = insb(insb(k, bl), bh);
    unsigned i00 = b00, i01 = b00 | ml, i10 = b00 | mh, i11 = b00 | ml | mh;
    float2 p, q, np, nq;
    // {00,11}: [[cd,-i sd],[-i sd,cd]] then *(cc - i sc)
    p = S[i00]; q = S[i11];
    np = make_float2(cd * p.x + sd * q.y, cd * p.y - sd * q.x);
    nq = make_float2(sd * p.y + cd * q.x, -sd * p.x + cd * q.y);
    S[i00] = make_float2(np.x * cc + np.y * sc, np.y * cc - np.x * sc);
    S[i11] = make_float2(nq.x * cc + nq.y * sc, nq.y * cc - nq.x * sc);
    // {01,10}: [[cs,-i ss],[-i ss,cs]] then *(cc + i sc)
    p = S[i01]; q = S[i10];
    np = make_float2(cs * p.x + ss * q.y, cs * p.y - ss * q.x);
    nq = make_float2(ss * p.y + cs * q.x, -ss * p.x + cs * q.y);
    S[i01] = make_float2(np.x * cc - np.y * sc, np.y * cc + np.x * sc);
    S[i10] = make_float2(nq.x * cc - nq.y * sc, nq.y * cc + nq.x * sc);
  }
}

// CCRX: controls cb1, cb2 set -> RX(pi/4) on tb
__device__ __forceinline__ void applyCCRX(float2* __restrict__ S, int cb1, int cb2, int tb)
{
  int s0 = cb1, s1 = cb2, s2 = tb, t;
  if (s0 > s1) { t = s0; s0 = s1; s1 = t; }
  if (s1 > s2) { t = s1; s1 = s2; s2 = t; }
  if (s0 > s1) { t = s0; s0 = s1; s1 = t; }
  unsigned m1 = 1u << cb1, m2 = 1u << cb2, mt = 1u << tb;
  const float cc = 0.92387953f;  // cos(pi/8)
  const float ss = 0.38268343f;  // sin(pi/8)
  for (unsigned k = threadIdx.x; k < 8192u; k += BLOCK) {
    unsigned base = insb(insb(insb(k, s0), s1), s2);
    unsigned i0 = base | m1 | m2;   // target bit = 0
    unsigned i1 = i0 | mt;
    float2 p = S[i0], q = S[i1];
    S[i0] = make_float2(cc * p.x + ss * q.y, cc * p.y - ss * q.x);
    S[i1] = make_float2(ss * p.y + cc * q.x, -ss * p.x + cc * q.y);
  }
}

// SWAP(a,r): exchange amplitudes with (b_a,b_r) = (1,0) <-> (0,1)
__device__ __forceinline__ void applySWAP(float2* __restrict__ S, int a, int r)
{
  int s0 = a < r ? a : r, s1 = a < r ? r : a;
  unsigned ma = 1u << a, mr = 1u << r;
  for (unsigned k = threadIdx.x; k < 16384u; k += BLOCK) {
    unsigned base = insb(insb(k, s0), s1);
    unsigned iA = base | ma, iR = base | mr;
    float2 tmp = S[iA]; S[iA] = S[iR]; S[iR] = tmp;
  }
}

// ---------------------------------------------------------------------------
__global__ __launch_bounds__(BLOCK, 1) void qpie_kernel(
    const float* __restrict__ x, const float* __restrict__ w,
    const float* __restrict__ pw, float* __restrict__ out,
    float2* __restrict__ ws)
{
  const int batch = blockIdx.x;
  float2* S = ws + (size_t)batch * NSTATE;
  const unsigned tid = threadIdx.x;

  __shared__ float t0[256], t1[256];
  __shared__ float evs[NQ];

  // |0..0> then H on qubits 0..7  ->  amp = 1/16 for indices with bits 8..15 == 0
  for (unsigned i = tid; i < NSTATE; i += BLOCK)
    S[i] = (i < 256u) ? make_float2(0.0625f, 0.f) : make_float2(0.f, 0.f);
  wgsync();

  const float r = 0.70710678f;
  int rec = NQ + NA;  // 10

  for (int layer = 0; layer < NL; ++layer) {
    // ---- input rotations (RY even layer / RX odd layer) ----
    for (int q = 0; q < NQ; ++q) {
      float t = x[batch * NQ + q];
      float ch = __cosf(0.5f * t), sh = __sinf(0.5f * t);
      if ((layer & 1) == 0)
        apply1q(S, q, ch, 0.f, -sh, 0.f, sh, 0.f, ch, 0.f);           // RY
      else
        apply1q(S, q, ch, 0.f, 0.f, -sh, 0.f, -sh, ch, 0.f);          // RX
      wgsync();
    }
    // ---- H on ancillas (bits 8,9) ----
    apply1q(S, 8, r, 0.f, r, 0.f, r, 0.f, -r, 0.f); wgsync();
    apply1q(S, 9, r, 0.f, r, 0.f, r, 0.f, -r, 0.f); wgsync();

    // ---- all 16 CRZ gates fused into one diagonal phase pass ----
    const float* pw0 = pw + (layer * NA + 0) * NQ;
    const float* pw1 = pw + (layer * NA + 1) * NQ;
    for (unsigned j = tid; j < 256u; j += BLOCK) {
      float a0 = 0.f, a1 = 0.f;
      for (int q = 0; q < NQ; ++q)
        if ((j >> q) & 1u) { a0 += pw0[q]; a1 += pw1[q]; }
      t0[j] = a0; t1[j] = a1;
    }
    wgsync();
    for (unsigned i = tid; i < NSTATE; i += BLOCK) {
      float sA = t0[i & 255u], sB = t1[i & 255u];
      float phi = 0.5f * ((((i >> 8) & 1u) ? sA : -sA) +
                          (((i >> 9) & 1u) ? sB : -sB));
      float sp, cp; __sincosf(phi, &sp, &cp);
      float2 v = S[i];
      S[i] = make_float2(v.x * cp - v.y * sp, v.x * sp + v.y * cp);
    }
    wgsync();

    // ---- (odd layers: H on ancilla) then SWAP ancilla -> recorder ----
    int rec0 = rec, rec1 = rec + 1;
    if (layer & 1) { apply1q(S, 8, r, 0.f, r, 0.f, r, 0.f, -r, 0.f); wgsync(); }
    applySWAP(S, 8, rec0); wgsync();
    if (layer & 1) { apply1q(S, 9, r, 0.f, r, 0.f, r, 0.f, -r, 0.f); wgsync(); }
    applySWAP(S, 9, rec1); wgsync();
    rec += 2;

    // ---- entangling pairs (0,1)(2,3)(4,5)(6,7) + CCRX on first two ----
    const float* wl = w + layer * 48;
    for (int i = 0; i < 4; ++i) {
      int q1 = 2 * i, q2 = 2 * i + 1;
      applyXXYYZZ(S, q1, q2, wl[3 * i], wl[3 * i + 1], wl[3 * i + 2]); wgsync();
      if (i < NA) { applyCCRX(S, (i == 0 ? rec0 : rec1), q1, q2); wgsync(); }
    }
    // ---- offset pairs (1,2)(3,4)(5,6) ----
    for (int i = 0; i < 3; ++i) {
      int q1 = 2 * i + 1, q2 = 2 * i + 2, idx = 4 + i;
      applyXXYYZZ(S, q1, q2, wl[3 * idx], wl[3 * idx + 1], wl[3 * idx + 2]); wgsync();
    }
    // ---- per-qubit RX,RY,RZ fused into one dense 2x2 gate: U = Rz*Ry*Rx ----
    const float* vw = wl + 24;
    for (int q = 0; q < NQ; ++q) {
      float al = vw[3 * q], be = vw[3 * q + 1], ga = vw[3 * q + 2];
      float ca = __cosf(0.5f * al), sa = __sinf(0.5f * al);
      float cb = __cosf(0.5f * be), sb = __sinf(0.5f * be);
      float cg = __cosf(0.5f * ga), sg = __sinf(0.5f * ga);
      float m00r = cb * ca,  m00i = sb * sa;
      float m01r = -sb * ca, m01i = -cb * sa;
      float m10r = sb * ca,  m10i = -cb * sa;
      float m11r = cb * ca,  m11i = -sb * sa;
      // row0 *= e^{-i g/2}, row1 *= e^{+i g/2}
      float u00r = m00r * cg + m00i * sg, u00i = m00i * cg - m00r * sg;
      float u01r = m01r * cg + m01i * sg, u01i = m01i * cg - m01r * sg;
      float u10r = m10r * cg - m10i * sg, u10i = m10i * cg + m10r * sg;
      float u11r = m11r * cg - m11i * sg, u11i = m11i * cg + m11r * sg;
      apply1q(S, q, u00r, u00i, u01r, u01i, u10r, u10i, u11r, u11i);
      wgsync();
    }
  }

  // ---- expectations: ev[q] = sum |amp|^2 * (1 - 2*bit_q) ----
  float acc[NQ];
#pragma unroll
  for (int q = 0; q < NQ; ++q) acc[q] = 0.f;
  for (unsigned i = tid; i < NSTATE; i += BLOCK) {
    float2 v = S[i];
    float p = v.x * v.x + v.y * v.y;
#pragma unroll
    for (int q = 0; q < NQ; ++q) acc[q] += ((i >> q) & 1u) ? -p : p;
  }
  if (tid < NQ) evs[tid] = 0.f;
  __syncthreads();
#pragma unroll
  for (int q = 0; q < NQ; ++q) atomicAdd(&evs[q], acc[q]);
  __syncthreads();
  if (tid < NQ) out[batch * NQ + tid] = evs[tid];
}

// ---------------------------------------------------------------------------
extern "C" void kernel_launch(void* const* d_in, const int* in_sizes, int n_in,
                              void* d_out, int out_size, void* d_ws, size_t ws_size,
                              hipStream_t stream) {
  (void)n_in; (void)out_size; (void)ws_size;
  const float* x  = (const float*)d_in[0];   // (16, 8)
  const float* w  = (const float*)d_in[1];   // (3, 48)
  const float* pw = (const float*)d_in[2];   // (3, 2, 8)
  float* out = (float*)d_out;                // (16, 8)
  int nbatch = in_sizes[0] / NQ;             // 16; needs nbatch*512KB of d_ws
  qpie_kernel<<<nbatch, BLOCK, 0, stream>>>(x, w, pw, out, (float2*)d_ws);
}